// PinGNN_87050397155920
// MI455X (gfx1250) — compile-verified
//
#include <hip/hip_runtime.h>

// ---------------------------------------------------------------------------
// PinGNN on MI455X (gfx1250): fused gather + f32 WMMA GEMM + atomic scatter,
// with CDNA5 async data movement (TDM weight staging + async gather-to-LDS).
// Reference shapes: N=50000 nodes, E=800000 edges, C=64 feats, Ce=16 edge feats.
// ---------------------------------------------------------------------------

#define NNODES 50000
#define NEDGES 800000
#define CF     64                 // C_IN == C_HID == C_NEIGH == C_OUT
#define CE     16
#define KEDGE  (2 * CF + CE)      // 144  (concat [x_dst, x_src, e_attr])
#define KNODE  (2 * CF)           // 128  (concat [x, agg])
#define NGRAPH 64

typedef float v2f __attribute__((ext_vector_type(2)));
typedef float v8f __attribute__((ext_vector_type(8)));

// ---- Feature detection for CDNA5 async paths (fallback = sync, known-good) --
#if defined(__has_builtin)
#if __has_builtin(__builtin_amdgcn_global_load_async_to_lds_b32) && \
    __has_builtin(__builtin_amdgcn_s_wait_asynccnt)
#define HAVE_ASYNC_LDS 1
#endif
#if __has_builtin(__builtin_amdgcn_tensor_load_to_lds) && \
    __has_builtin(__builtin_amdgcn_s_wait_tensorcnt)
#define HAVE_TDM 1
#endif
#endif

#if defined(HAVE_ASYNC_LDS)
// Per-lane gather straight into LDS (ASYNCcnt-tracked, no VGPR round trip).
// Builtin signature (from hipcc diagnostic): (global int*, local int*, Ii, Ii).
__device__ __forceinline__ void async_ld_b32(const float* g, float* l) {
  __builtin_amdgcn_global_load_async_to_lds_b32(
      (__attribute__((address_space(1))) int*)g,
      (__attribute__((address_space(3))) int*)l, /*offset=*/0, /*cpol=*/0);
}
#endif

#if defined(HAVE_TDM)
typedef unsigned int u32x4 __attribute__((ext_vector_type(4)));
typedef int          i32x4 __attribute__((ext_vector_type(4)));
typedef int          i32x8 __attribute__((ext_vector_type(8)));

// TDM: DMA a contiguous block of `ndw` dwords (ndw < 65536) global -> LDS.
// D# per CDNA5 ISA ch.8: 1-row 2D tile, data_size=4B, groups 2/3 unused.
// 6-arg builtin form (this toolchain): (u32x4, i32x8, i32x4, i32x4, i32x8, cpol).
__device__ __forceinline__ void tdm_load_block(const float* gptr, float* lptr,
                                               unsigned ndw) {
  const unsigned long long ga = (unsigned long long)gptr;
  const unsigned la =
      (unsigned)(unsigned long long)(__attribute__((address_space(3))) char*)lptr;
  u32x4 g0;
  g0.x = 1u;                                               // count=1 (valid D#)
  g0.y = la;                                               // lds_addr
  g0.z = (unsigned)ga;                                     // global_addr[31:0]
  g0.w = ((unsigned)(ga >> 32) & 0x01FFFFFFu) | (2u << 30); // addr[56:32]|type=2
  i32x8 g1;
  g1[0] = (int)(2u << 16);                        // data_size = 4 bytes
  g1[1] = (int)((ndw & 0xFFFFu) << 16);           // tensor_dim0[15:0]
  g1[2] = (int)(((ndw >> 16) & 0xFFFFu) | (1u << 16)); // dim0 hi | tensor_dim1=1
  g1[3] = (int)((ndw & 0xFFFFu) << 16);           // tile_dim0 = ndw
  g1[4] = 1;                                      // tile_dim1 = 1, tile_dim2 = 0
  g1[5] = (int)ndw;                               // tensor_dim0_stride lo
  g1[6] = 0;
  g1[7] = 0;
  const i32x4 gz4 = {0, 0, 0, 0};
  const i32x8 gz8 = {0, 0, 0, 0, 0, 0, 0, 0};
  __builtin_amdgcn_tensor_load_to_lds(g0, g1, gz4, gz4, gz8, /*cpol=*/0);
}
#endif

__device__ __forceinline__ void wait_async_gather() {
#if defined(HAVE_ASYNC_LDS)
  __builtin_amdgcn_s_wait_asynccnt(0);
#endif
}

// D = A(16x4 f32) * B(4x16 f32) + C(16x16 f32), wave32, one matrix per wave.
__device__ __forceinline__ v8f wmma4_f32(v2f a, v2f b, v8f c) {
  return __builtin_amdgcn_wmma_f32_16x16x4_f32(
      /*neg_a=*/false, a, /*neg_b=*/false, b,
      /*c_mod=*/(short)0, c, /*reuse_a=*/false, /*reuse_b=*/false);
}

// ---------------------------------------------------------------------------
// Edge message kernel: per wave, a 16-edge tile.
//   A[16 x 144] = [x[dst] | x[src] | edge_attr]  (gathered, L2-resident x)
//   msg[16 x 64] = A @ W_eff + b_eff
//   agg[dst] += msg   (global f32 atomics, L2 resident)
// ---------------------------------------------------------------------------
#define EDGE_WAVES 2
#define ASTR 145   // 16-row LDS tile stride: n*145 mod 64 = n*17 -> conflict free

__global__ __launch_bounds__(EDGE_WAVES * 32)
void edge_msg_kernel(const float* __restrict__ xin,
                     const float* __restrict__ eattr,
                     const float* __restrict__ Weff,
                     const float* __restrict__ beff,
                     const int*   __restrict__ srcIdx,
                     const int*   __restrict__ dstIdx,
                     float*       __restrict__ agg)
{
  __shared__ float sW[KEDGE * CF];                // 36864 B
  __shared__ float sA[EDGE_WAVES][16 * ASTR];     // 2 x 9280 B

  const int tid  = threadIdx.x;
  const int wave = tid >> 5;
  const int lane = tid & 31;

  // Stage W_eff (144x64) once per block: TDM DMA if available, else VMEM.
#if defined(HAVE_TDM)
  if (wave == 0) tdm_load_block(Weff, &sW[0], KEDGE * CF);
#else
  for (int i = tid; i < KEDGE * CF; i += EDGE_WAVES * 32) sW[i] = Weff[i];
#endif

  // Prefetch next tile's streaming edge_attr.
  const int ebase = (blockIdx.x * EDGE_WAVES + wave) * 16;
  {
    int pe = ebase + 16 * EDGE_WAVES;
    if (pe >= NEDGES) pe = NEDGES - 1;
    __builtin_prefetch(&eattr[(size_t)pe * CE], 0, 0);
  }

  // Stage this wave's 16x144 A tile (gather -> LDS).
  for (int r = 0; r < 16; ++r) {
    const int e = ebase + r;
    const int d = dstIdx[e];
    const int s = srcIdx[e];
    float* rowp = &sA[wave][r * ASTR];
    for (int c = lane; c < KEDGE; c += 32) {
      const float* gp;
      if      (c < CF)      gp = &xin[d * CF + c];
      else if (c < 2 * CF)  gp = &xin[s * CF + (c - CF)];
      else                  gp = &eattr[e * CE + (c - 2 * CF)];
#if defined(HAVE_ASYNC_LDS)
      async_ld_b32(gp, rowp + c);
#else
      rowp[c] = *gp;
#endif
    }
  }
  wait_async_gather();
#if defined(HAVE_TDM)
  if (wave == 0) __builtin_amdgcn_s_wait_tensorcnt(0);
#endif
  __syncthreads();

  // WMMA: 36 K-steps (K=144) x 4 N-tiles (N=64).
  const int half = lane >> 4;
  const int n    = lane & 15;
  const float* Arow = &sA[wave][n * ASTR];

  v8f acc[4] = {};
  for (int k0 = 0; k0 < KEDGE; k0 += 4) {
    const int ka = k0 + 2 * half;           // K index this lane-half owns
    v2f a;
    a.x = Arow[ka];
    a.y = Arow[ka + 1];
    const float* Wk0 = &sW[ka * CF];
    const float* Wk1 = &sW[(ka + 1) * CF];
#pragma unroll
    for (int t = 0; t < 4; ++t) {
      v2f b;
      b.x = Wk0[t * 16 + n];
      b.y = Wk1[t * 16 + n];
      acc[t] = wmma4_f32(a, b, acc[t]);
    }
  }

  // Bias + atomic scatter into agg[dst].
  int drow[8];
#pragma unroll
  for (int v = 0; v < 8; ++v) drow[v] = dstIdx[ebase + v + 8 * half];

#pragma unroll
  for (int t = 0; t < 4; ++t) {
    const int col = t * 16 + n;
    const float bb = beff[col];
#pragma unroll
    for (int v = 0; v < 8; ++v) {
      atomicAdd(&agg[drow[v] * CF + col], acc[t][v] + bb);
    }
  }
}

// ---------------------------------------------------------------------------
// Node update kernel: out = [x | agg] @ W_out + b_out (+ optional ReLU).
// One wave per 16-node tile.
// ---------------------------------------------------------------------------
#define NSTR 129

__global__ __launch_bounds__(32)
void node_out_kernel(const float* __restrict__ xin,
                     const float* __restrict__ agg,
                     const float* __restrict__ Wout,
                     const float* __restrict__ bout,
                     float*       __restrict__ out,
                     const int applyRelu)
{
  __shared__ float sW[KNODE * CF];   // 32768 B
  __shared__ float sA[16 * NSTR];    //  8256 B

  const int lane = threadIdx.x;

#if defined(HAVE_TDM)
  tdm_load_block(Wout, &sW[0], KNODE * CF);
#else
  for (int i = lane; i < KNODE * CF; i += 32) sW[i] = Wout[i];
#endif

  const int nbase = blockIdx.x * 16;
  for (int r = 0; r < 16; ++r) {
    const int node = nbase + r;
    float* rowp = &sA[r * NSTR];
    for (int c = lane; c < KNODE; c += 32) {
      const float* gp = (c < CF) ? &xin[node * CF + c]
                                 : &agg[node * CF + (c - CF)];
#if defined(HAVE_ASYNC_LDS)
      async_ld_b32(gp, rowp + c);
#else
      rowp[c] = *gp;
#endif
    }
  }
  wait_async_gather();
#if defined(HAVE_TDM)
  __builtin_amdgcn_s_wait_tensorcnt(0);
#endif
  __syncthreads();

  const int half = lane >> 4;
  const int n    = lane & 15;
  const float* Arow = &sA[n * NSTR];

  v8f acc[4] = {};
  for (int k0 = 0; k0 < KNODE; k0 += 4) {
    const int ka = k0 + 2 * half;
    v2f a;
    a.x = Arow[ka];
    a.y = Arow[ka + 1];
    const float* Wk0 = &sW[ka * CF];
    const float* Wk1 = &sW[(ka + 1) * CF];
#pragma unroll
    for (int t = 0; t < 4; ++t) {
      v2f b;
      b.x = Wk0[t * 16 + n];
      b.y = Wk1[t * 16 + n];
      acc[t] = wmma4_f32(a, b, acc[t]);
    }
  }

#pragma unroll
  for (int t = 0; t < 4; ++t) {
    const int col = t * 16 + n;
    const float bb = bout[col];
#pragma unroll
    for (int v = 0; v < 8; ++v) {
      const int node = nbase + v + 8 * half;
      float val = acc[t][v] + bb;
      if (applyRelu) val = fmaxf(val, 0.0f);
      out[node * CF + col] = val;
    }
  }
}

// ---------------------------------------------------------------------------
// Helpers: zero-fill, segment pooling.
// ---------------------------------------------------------------------------
__global__ void fill_f32(float* __restrict__ p, long long nElem, float val) {
  long long i = (long long)blockIdx.x * blockDim.x + threadIdx.x;
  long long stride = (long long)gridDim.x * blockDim.x;
  for (; i < nElem; i += stride) p[i] = val;
}

__global__ void pool_sum_kernel(const float* __restrict__ h2,
                                const int* __restrict__ gidx,
                                float* __restrict__ sums) {
  const int i = blockIdx.x * blockDim.x + threadIdx.x;
  if (i < NNODES * CF) {
    const int node = i >> 6;
    const int col  = i & 63;
    const int g    = gidx[node];
    atomicAdd(&sums[g * CF + col], h2[i]);
  }
}

__global__ void pool_cnt_kernel(const int* __restrict__ gidx,
                                float* __restrict__ cnts) {
  const int node = blockIdx.x * blockDim.x + threadIdx.x;
  if (node < NNODES) atomicAdd(&cnts[gidx[node]], 1.0f);
}

__global__ void pool_div_kernel(const float* __restrict__ sums,
                                const float* __restrict__ cnts,
                                float* __restrict__ out) {
  const int i = blockIdx.x * blockDim.x + threadIdx.x;
  if (i < NGRAPH * CF) out[i] = sums[i] / fmaxf(cnts[i >> 6], 1.0f);
}

// ---------------------------------------------------------------------------
// Launch sequence (graph-capture safe; deterministic; all on `stream`).
// ---------------------------------------------------------------------------
extern "C" void kernel_launch(void* const* d_in, const int* in_sizes, int n_in,
                              void* d_out, int out_size, void* d_ws, size_t ws_size,
                              hipStream_t stream) {
  const float* x    = (const float*)d_in[0];
  const float* ea   = (const float*)d_in[1];
  const float* W1e  = (const float*)d_in[2];
  const float* b1e  = (const float*)d_in[3];
  const float* W1o  = (const float*)d_in[4];
  const float* b1o  = (const float*)d_in[5];
  const float* W2e  = (const float*)d_in[6];
  const float* b2e  = (const float*)d_in[7];
  const float* W2o  = (const float*)d_in[8];
  const float* b2o  = (const float*)d_in[9];
  const int*   eidx = (const int*)d_in[10];   // [2, E] flat: row0 = src, row1 = dst
  const int*   gidx = (const int*)d_in[11];
  const int* srcIdx = eidx;
  const int* dstIdx = eidx + NEDGES;

  // Workspace layout (f32): agg | h | h2 | sums | counts
  float* agg  = (float*)d_ws;
  float* h    = agg  + (size_t)NNODES * CF;
  float* h2   = h    + (size_t)NNODES * CF;
  float* sums = h2   + (size_t)NNODES * CF;
  float* cnts = sums + (size_t)NGRAPH * CF;   // contiguous with sums

  const long long aggN = (long long)NNODES * CF;
  const int edgeBlocks = NEDGES / (16 * EDGE_WAVES);  // 25000 (exact)
  const int nodeBlocks = NNODES / 16;                 // 3125  (exact)

  // ---- Layer 1 ----
  fill_f32<<<1024, 256, 0, stream>>>(agg, aggN, 0.0f);
  edge_msg_kernel<<<edgeBlocks, EDGE_WAVES * 32, 0, stream>>>(
      x, ea, W1e, b1e, srcIdx, dstIdx, agg);
  node_out_kernel<<<nodeBlocks, 32, 0, stream>>>(x, agg, W1o, b1o, h, /*relu=*/1);

  // ---- Layer 2 ----
  fill_f32<<<1024, 256, 0, stream>>>(agg, aggN, 0.0f);
  edge_msg_kernel<<<edgeBlocks, EDGE_WAVES * 32, 0, stream>>>(
      h, ea, W2e, b2e, srcIdx, dstIdx, agg);
  node_out_kernel<<<nodeBlocks, 32, 0, stream>>>(h, agg, W2o, b2o, h2, /*relu=*/0);

  // ---- Global mean pool ----
  fill_f32<<<32, 256, 0, stream>>>(sums, (long long)(NGRAPH * CF + NGRAPH), 0.0f);
  pool_sum_kernel<<<(NNODES * CF + 255) / 256, 256, 0, stream>>>(h2, gidx, sums);
  pool_cnt_kernel<<<(NNODES + 255) / 256, 256, 0, stream>>>(gidx, cnts);
  pool_div_kernel<<<(NGRAPH * CF + 255) / 256, 256, 0, stream>>>(sums, cnts,
                                                                 (float*)d_out);
}